// PseudoTokenGridEncoder_54683523612696
// MI455X (gfx1250) — compile-verified
//
#include <hip/hip_runtime.h>
#include <hip/hip_bf16.h>

// ---------------------------------------------------------------------------
// PseudoTokenGridEncoder for MI455X (gfx1250, wave32, WMMA + async-LDS).
// Factored-attention formulation: bandwidth-bound (~130MB @ 23.3TB/s), so we
// keep full fp32 and use V_WMMA_F32_16X16X4_F32 for the three real GEMMs.
// Token gather / G staging use GLOBAL_LOAD_ASYNC_TO_LDS_B128 when available.
// ---------------------------------------------------------------------------

typedef __attribute__((ext_vector_type(2))) float v2f;
typedef __attribute__((ext_vector_type(8))) float v8f;
typedef __attribute__((ext_vector_type(4))) int   v4i;

#define B_   8
#define U_   8192
#define E_   256
#define N_   32          // grid points per dim
#define S_   1024        // N_*N_ cells
#define MP_  40
#define H_   8
#define HD_  32

// ---- gfx1250 async global->LDS path (guarded; sync fallback otherwise) ----
// Builtin signature (from hipcc diagnostic): param0 = v4i* in AS1 (__device__),
// param1 = LDS-side v4i*, then two integer immediates (offset, cpol).
#if defined(__gfx1250__) && __has_builtin(__builtin_amdgcn_global_load_async_to_lds_b128)
#define HAVE_ASYNC_LDS 1
// AS1 (global): flat address == global address numerically -> integer cast.
#define GPTR(p) ((__attribute__((address_space(1))) v4i*)(unsigned long long)(p))
// AS3 (LDS): explicit overlap addrspace cast from generic.
#define LPTR(p) ((__attribute__((address_space(3))) v4i*)(p))
#if __has_builtin(__builtin_amdgcn_s_wait_asynccnt)
#define WAIT_ASYNC() __builtin_amdgcn_s_wait_asynccnt(0)
#else
#define WAIT_ASYNC() asm volatile("s_wait_asynccnt 0x0" ::: "memory")
#endif
#else
#define HAVE_ASYNC_LDS 0
#endif

// ---------------------------------------------------------------- cell ids
__global__ void cells_kernel(const float* __restrict__ xc_off,
                             int* __restrict__ cellId) {
  int t = blockIdx.x * blockDim.x + threadIdx.x;       // 0 .. B*U-1
  if (t >= B_ * U_) return;
  float x = xc_off[2 * t + 0];
  float y = xc_off[2 * t + 1];
  int r = (int)__builtin_rintf(x * (float)(N_ - 1));   // round-half-even == jnp.round
  int c = (int)__builtin_rintf(y * (float)(N_ - 1));
  r = min(max(r, 0), N_ - 1);
  c = min(max(c, 0), N_ - 1);
  cellId[t] = r * N_ + c;
}

// ------------------------------------------------- deterministic bucketize
// One block per batch, one thread per cell. Scans points in u-order through
// an LDS staging buffer -> exact reference cumcount ordering, no atomics.
__global__ __launch_bounds__(1024) void bucketize_kernel(
    const int* __restrict__ cellId,
    int* __restrict__ counts,        // [B*S]
    int* __restrict__ idxl) {        // [B*S, MP-1]
  __shared__ int sCell[1024];
  const int b = blockIdx.x;
  const int c = threadIdx.x;         // my cell
  const int* cb = cellId + (size_t)b * U_;
  const int bucket = b * S_ + c;
  int cnt = 0;
  for (int base = 0; base < U_; base += 1024) {
#if HAVE_ASYNC_LDS
    if (threadIdx.x < 256) {         // waves 0-7: 4 ints per lane, b128
      __builtin_amdgcn_global_load_async_to_lds_b128(
          GPTR(cb + base + threadIdx.x * 4), LPTR(&sCell[threadIdx.x * 4]),
          0, 0);
      WAIT_ASYNC();                  // per-wave drain before the barrier
    }
#else
    sCell[threadIdx.x] = cb[base + threadIdx.x];
#endif
    __syncthreads();
    for (int j = 0; j < 1024; ++j) {
      if (sCell[j] == c) {
        if (cnt < MP_ - 1) idxl[(size_t)bucket * (MP_ - 1) + cnt] = base + j;
        cnt++;
      }
    }
    __syncthreads();
  }
  counts[bucket] = cnt;
}

// ------------------------------------------- fp32 WMMA GEMM, 16x64 per wave
// C[M,N] = A[M,K] @ B[K,N], row-major, M%16==0, N%64==0, K%4==0.
// Each wave owns a 16(M)x64(N) strip: one A fragment feeds 4 WMMAs per
// k-step (10 operand loads / 4 WMMAs instead of 16/4).
// Fragment layouts per CDNA5 ISA 7.12.2:
//   A 16x4 : lanes 0-15 rows M, VGPRs K=0,1 ; lanes 16-31 rows M, K=2,3
//   B 4x16 : lanes 0-15 cols N, rows K=0,1 ; lanes 16-31 cols N, K=2,3
//   C      : VGPR j -> M = j (+8 for upper half), N = lane%16
__global__ __launch_bounds__(32) void gemm16x64_kernel(
    const float* __restrict__ A, const float* __restrict__ Bm,
    float* __restrict__ C, int M, int N, int K) {
  const int ntn  = N >> 6;                     // 64-wide strips
  const int tm   = blockIdx.x / ntn;
  const int tnS  = (blockIdx.x % ntn) << 6;    // strip col base
  const int lane = threadIdx.x;
  const int half = lane >> 4;
  const int l    = lane & 15;
  const float* Arow  = A  + (size_t)(tm * 16 + l) * K + 2 * half;
  const float* Bbase = Bm + (size_t)(2 * half) * N + tnS + l;
  v8f acc0 = {}, acc1 = {}, acc2 = {}, acc3 = {};
  for (int k0 = 0; k0 < K; k0 += 4) {
    const size_t r0 = (size_t)k0 * N, r1 = (size_t)(k0 + 1) * N;
    v2f a, b0, b1, b2, b3;
    a.x  = Arow[k0];          a.y  = Arow[k0 + 1];
    b0.x = Bbase[r0 +  0];    b0.y = Bbase[r1 +  0];
    b1.x = Bbase[r0 + 16];    b1.y = Bbase[r1 + 16];
    b2.x = Bbase[r0 + 32];    b2.y = Bbase[r1 + 32];
    b3.x = Bbase[r0 + 48];    b3.y = Bbase[r1 + 48];
    acc0 = __builtin_amdgcn_wmma_f32_16x16x4_f32(false, a, false, b0,
                                                 (short)0, acc0, false, false);
    acc1 = __builtin_amdgcn_wmma_f32_16x16x4_f32(false, a, false, b1,
                                                 (short)0, acc1, false, false);
    acc2 = __builtin_amdgcn_wmma_f32_16x16x4_f32(false, a, false, b2,
                                                 (short)0, acc2, false, false);
    acc3 = __builtin_amdgcn_wmma_f32_16x16x4_f32(false, a, false, b3,
                                                 (short)0, acc3, false, false);
  }
  float* Cout = C + (size_t)(tm * 16 + half * 8) * N + tnS + l;
#pragma unroll
  for (int j = 0; j < 8; ++j) {
    Cout[(size_t)j * N +  0] = acc0[j];
    Cout[(size_t)j * N + 16] = acc1[j];
    Cout[(size_t)j * N + 32] = acc2[j];
    Cout[(size_t)j * N + 48] = acc3[j];
  }
}

// ------------------------------------------------ G precompute (WMMA, K=32)
// G[c,h,e] = sum_{d<32} Q[c, h*32+d] * Wk[e, h*32+d]
// Per head: GEMM M=S_, N=E_, K=32 with B = Wk-transposed head slice.
// blockIdx = ((h*64 + tm)*16 + tn)
__global__ __launch_bounds__(32) void gscore_kernel(
    const float* __restrict__ Q, const float* __restrict__ Wk,
    float* __restrict__ G) {
  const int tn  = blockIdx.x & 15;
  const int tmh = blockIdx.x >> 4;
  const int tm  = tmh & 63;
  const int h   = tmh >> 6;
  const int lane = threadIdx.x;
  const int half = lane >> 4;
  const int l    = lane & 15;
  const float* Arow = Q  + (size_t)(tm * 16 + l) * E_ + h * HD_ + 2 * half;
  const float* Brow = Wk + (size_t)(tn * 16 + l) * E_ + h * HD_ + 2 * half;
  v8f acc = {};
#pragma unroll
  for (int k0 = 0; k0 < HD_; k0 += 4) {
    v2f a, b;
    a.x = Arow[k0];     a.y = Arow[k0 + 1];
    b.x = Brow[k0];     b.y = Brow[k0 + 1];
    acc = __builtin_amdgcn_wmma_f32_16x16x4_f32(false, a, false, b,
                                                (short)0, acc, false, false);
  }
  // C row c = tm*16 + half*8 + j, col e = tn*16 + l -> G[(c*H + h)*E + e]
  float* Gout = G + ((size_t)(tm * 16 + half * 8) * H_ + h) * E_ + tn * 16 + l;
#pragma unroll
  for (int j = 0; j < 8; ++j) Gout[(size_t)j * H_ * E_] = acc[j];
}

// -------------------------------------------------- fused bucket attention
// One 256-thread block per (b,cell) bucket. Gathers <=40 tokens to LDS
// (async global->LDS b128 when available), scores against staged G[cell],
// softmax per head, token-space weighted sum w[h,e], folds through Wv.
__global__ __launch_bounds__(256) void attn_kernel(
    const float* __restrict__ zc_off, const float* __restrict__ zc_on,
    const float* __restrict__ fake,  const float* __restrict__ Wv,
    const float* __restrict__ G,     const int* __restrict__ counts,
    const int* __restrict__ idxl,    const int* __restrict__ ignore_flag,
    float* __restrict__ attnO) {
  __shared__ float sT[MP_][E_];       // raw tokens           40 KB
  __shared__ float sG[H_][E_];        // per-cell score vecs   8 KB
  __shared__ float sW[H_][E_];        // weighted tokens       8 KB
  __shared__ float sScore[H_][MP_];
  __shared__ int   sIdx[MP_ - 1];
  __shared__ int   sMeta;

  const int bucket = blockIdx.x;
  const int b    = bucket / S_;
  const int cell = bucket - b * S_;
  const int tid  = threadIdx.x;

  // warm Wv rows for the tail stage (global_prefetch_b8, speculative)
  __builtin_prefetch(Wv + (size_t)tid * E_, 0, 0);

  if (tid == 0) sMeta = min(counts[bucket], MP_ - 1);
  if (tid < MP_ - 1) sIdx[tid] = idxl[(size_t)bucket * (MP_ - 1) + tid];

  // stage G[cell] (8x256 f32 = 512 x b128 chunks; 2 chunks per lane)
  {
    const float* gsrc = G + (size_t)cell * H_ * E_;
#if HAVE_ASYNC_LDS
    __builtin_amdgcn_global_load_async_to_lds_b128(
        GPTR(gsrc + tid * 4), LPTR(&sG[0][0] + tid * 4), 0, 0);
    __builtin_amdgcn_global_load_async_to_lds_b128(
        GPTR(gsrc + (tid + 256) * 4), LPTR(&sG[0][0] + (tid + 256) * 4), 0, 0);
#else
    (&sG[0][0])[tid]       = gsrc[tid];
    (&sG[0][0])[tid + 256] = gsrc[tid + 256];
    (&sG[0][0])[tid + 512] = gsrc[tid + 512];
    (&sG[0][0])[tid + 768] = gsrc[tid + 768];
    (&sG[0][0])[tid + 1024] = gsrc[tid + 1024];
    (&sG[0][0])[tid + 1280] = gsrc[tid + 1280];
    (&sG[0][0])[tid + 1536] = gsrc[tid + 1536];
    (&sG[0][0])[tid + 1792] = gsrc[tid + 1792];
#endif
  }
  __syncthreads();                    // sIdx/sMeta visible (sG may be in flight)

  const int nOff = sMeta;
  const int nTok = nOff + 1;          // + on-grid/fake slot

  // gather tokens: 4 rows per pass, 64 lanes x 16B per row
  const float* last = ignore_flag[0] ? fake : (zc_on + (size_t)bucket * E_);
  {
    const int r  = tid >> 6;          // 0..3
    const int c4 = (tid & 63) << 2;   // float col, 16B aligned
    for (int p0 = 0; p0 < nTok; p0 += 4) {
      const int row = p0 + r;
      if (row < nTok) {
        const float* src =
            (row < nOff) ? (zc_off + ((size_t)b * U_ + sIdx[row]) * E_) : last;
#if HAVE_ASYNC_LDS
        __builtin_amdgcn_global_load_async_to_lds_b128(
            GPTR(src + c4), LPTR(&sT[row][c4]), 0, 0);
#else
        *(float4*)&sT[row][c4] = *(const float4*)(src + c4);
#endif
      }
    }
  }
#if HAVE_ASYNC_LDS
  WAIT_ASYNC();                       // drain this wave's sG + token loads
#endif
  __syncthreads();                    // all waves drained -> LDS coherent

  const float scale = 0.17677669529663687f;   // 1/sqrt(HD_)
  for (int t = tid; t < H_ * MP_; t += 256) {
    const int h = t / MP_, p = t - h * MP_;
    if (p < nTok) {
      float acc = 0.f;
      for (int e = 0; e < E_; ++e) acc += sT[p][e] * sG[h][e];
      sScore[h][p] = acc * scale;
    }
  }
  __syncthreads();

  if (tid < H_) {                     // tiny softmax over <=40 slots
    float mx = -3.4e38f;
    for (int p = 0; p < nTok; ++p) mx = fmaxf(mx, sScore[tid][p]);
    float sum = 0.f;
    for (int p = 0; p < nTok; ++p) {
      float v = expf(sScore[tid][p] - mx);
      sScore[tid][p] = v;
      sum += v;
    }
    const float inv = 1.0f / sum;
    for (int p = 0; p < nTok; ++p) sScore[tid][p] *= inv;
  }
  __syncthreads();

  {                                   // w[h,e] = sum_p attn[h,p] * T[p,e]
    const int e = tid;
    for (int h = 0; h < H_; ++h) {
      float acc = 0.f;
      for (int p = 0; p < nTok; ++p) acc += sScore[h][p] * sT[p][e];
      sW[h][e] = acc;
    }
  }
  __syncthreads();

  {                                   // out[o] = sum_e w[o/32,e] * Wv[e,o]
    const int o = tid;
    const int h = o >> 5;
    float acc = 0.f;
    for (int e = 0; e < E_; ++e) acc += sW[h][e] * Wv[(size_t)e * E_ + o];
    attnO[(size_t)bucket * E_ + o] = acc;
  }
}

// ------------------------------------------------------------ passthrough
__global__ void copy_kernel(const float* __restrict__ src,
                            float* __restrict__ dst, int n) {
  int t = blockIdx.x * blockDim.x + threadIdx.x;
  if (t < n) dst[t] = src[t];
}

// ---------------------------------------------------------------------------
extern "C" void kernel_launch(void* const* d_in, const int* in_sizes, int n_in,
                              void* d_out, int out_size, void* d_ws,
                              size_t ws_size, hipStream_t stream) {
  const float* xc_off  = (const float*)d_in[0];
  const float* xc_on   = (const float*)d_in[1];
  const float* zc_off  = (const float*)d_in[2];
  const float* zc_on   = (const float*)d_in[3];
  const float* latents = (const float*)d_in[4];
  const float* fake    = (const float*)d_in[5];
  const float* Wq      = (const float*)d_in[6];
  const float* Wk      = (const float*)d_in[7];
  const float* Wv      = (const float*)d_in[8];
  const float* Wo      = (const float*)d_in[9];
  const int*   ignore  = (const int*)d_in[10];

  // workspace layout (bytes)
  char* ws = (char*)d_ws;
  int*   cellId = (int*)(ws + 0);                    //  262,144
  int*   counts = (int*)(ws + 262144);               //   32,768
  int*   idxl   = (int*)(ws + 294912);               // 1,277,952
  float* Q      = (float*)(ws + 1572864);            // 1,048,576
  float* G      = (float*)(ws + 2621440);            // 8,388,608
  float* attnO  = (float*)(ws + 11010048);           // 8,388,608
  const size_t WS_NEED = 19398656;
  if (ws_size < WS_NEED) return;                     // deterministic bail

  float* out_xc = (float*)d_out;                     // 16384 floats
  float* out_z  = out_xc + B_ * S_ * 2;              // 2,097,152 floats

  // 1) cell ids + deterministic bucketize
  cells_kernel<<<(B_ * U_ + 255) / 256, 256, 0, stream>>>(xc_off, cellId);
  bucketize_kernel<<<B_, 1024, 0, stream>>>(cellId, counts, idxl);

  // 2) Q = latents @ Wq   (M=1024, N=256, K=256)  [WMMA 16x64/wave]
  gemm16x64_kernel<<<(S_ / 16) * (E_ / 64), 32, 0, stream>>>(latents, Wq, Q,
                                                             S_, E_, E_);
  // 3) G[c,h,:] = Wk-head @ q-head                 [WMMA]
  gscore_kernel<<<H_ * (S_ / 16) * (E_ / 16), 32, 0, stream>>>(Q, Wk, G);

  // 4) fused bucket attention -> attnO [B*S, E]
  attn_kernel<<<B_ * S_, 256, 0, stream>>>(zc_off, zc_on, fake, Wv, G, counts,
                                           idxl, ignore, attnO);

  // 5) outputs: xc passthrough + z = attnO @ Wo    [WMMA 16x64/wave]
  copy_kernel<<<(B_ * S_ * 2 + 255) / 256, 256, 0, stream>>>(xc_on, out_xc,
                                                             B_ * S_ * 2);
  gemm16x64_kernel<<<(B_ * S_ / 16) * (E_ / 64), 32, 0, stream>>>(
      attnO, Wo, out_z, B_ * S_, E_, E_);
}